// OrderedCapsuleLayer_10161892623159
// MI455X (gfx1250) — compile-verified
//
#include <hip/hip_runtime.h>
#include <hip/hip_bf16.h>

// ---- problem constants (from reference) ----
#define B_   16
#define I_   256
#define J_   128
#define DIN  512
#define D_   64
#define DOUT 512
#define MAXREL 128

typedef __attribute__((ext_vector_type(2))) float v2f;
typedef __attribute__((ext_vector_type(8))) float v8f;

// ---------------------------------------------------------------------------
// zero-init logits
// ---------------------------------------------------------------------------
__global__ void k_zero(float* __restrict__ p, int n) {
    int t = blockIdx.x * blockDim.x + threadIdx.x;
    if (t < n) p[t] = 0.0f;
}

// ---------------------------------------------------------------------------
// u = inputs_u @ W_in + b_in        [4096 x 512] * [512 x 64] -> [4096 x 64]
// grid.x = 256 M-tiles, block = 128 (4 waves, wave w = N-tile w)
// ---------------------------------------------------------------------------
__global__ void k_u_gemm(const float* __restrict__ A, const float* __restrict__ W,
                         const float* __restrict__ bias, float* __restrict__ U) {
    const int lane = threadIdx.x & 31;
    const int wave = threadIdx.x >> 5;
    const int m    = lane & 15;
    const int hi   = lane >> 4;          // selects K pair {0,1} / {2,3}
    const int row  = blockIdx.x * 16 + m;
    const int n0   = wave * 16;
    const int colB = lane & 15;

    v8f c = {};
    #pragma unroll 4
    for (int k = 0; k < DIN; k += 4) {
        const int kk = k + 2 * hi;
        v2f a = *(const v2f*)(A + row * DIN + kk);
        v2f b;
        b.x = W[kk * D_ + n0 + colB];
        b.y = W[(kk + 1) * D_ + n0 + colB];
        c = __builtin_amdgcn_wmma_f32_16x16x4_f32(false, a, false, b,
                                                  (short)0, c, false, false);
    }
    #pragma unroll
    for (int r = 0; r < 8; ++r) {
        const int Row = r + hi * 8;
        const int Col = lane & 15;
        U[(blockIdx.x * 16 + Row) * D_ + n0 + Col] = c[r] + bias[n0 + Col];
    }
}

// ---------------------------------------------------------------------------
// priors by diagonal: idx = max(j-i+128, 0); all (i,j) with same idx share
// RW[idx].  Block = (chunk, idx): RW[idx] staged in LDS once; 16 waves, wave t
// owns i = iLo + chunk*16 + t and computes the [16(batch) x 64] @ [64 x 64]
// product into 4 N-tiles.  idx==0 results (identical for all clipped j) go to
// T0 and are broadcast by k_priors_bcast.
// priors layout: [b][j][i][p]
// ---------------------------------------------------------------------------
__global__ void k_priors_diag(const float* __restrict__ U, const float* __restrict__ route,
                              float* __restrict__ priors, float* __restrict__ T0) {
    const int idx   = blockIdx.y;        // 0..255
    const int chunk = blockIdx.x;        // 0..7
    const int tid   = threadIdx.x;       // 0..511
    const int lane  = tid & 31;
    const int wavet = tid >> 5;          // 0..15 : M-tile == i-offset
    const int m     = lane & 15;         // batch row
    const int hi    = lane >> 4;
    const int colB  = lane & 15;

    __shared__ float Bs[D_ * D_];        // 16 KB: RW[idx]
    const float4* rw4 = (const float4*)(route + (size_t)idx * D_ * D_);
    float4* Bs4 = (float4*)Bs;
    Bs4[tid]       = rw4[tid];
    Bs4[tid + 512] = rw4[tid + 512];

    int iLo, cnt;
    if (idx == 0)        { iLo = 128;       cnt = 128; }
    else if (idx <= 128) { iLo = 128 - idx; cnt = 128; }
    else                 { iLo = 0;         cnt = 256 - idx; }
    const int  i    = iLo + chunk * 16 + wavet;
    const bool live = (chunk * 16 + wavet) < cnt;

    // preload this wave's A tile (16 batches x 64 k) into registers
    v2f a[16];
    if (live) {
        const float* arow = U + (m * I_ + i) * D_ + 2 * hi;
        #pragma unroll
        for (int s = 0; s < 16; ++s)
            a[s] = *(const v2f*)(arow + s * 4);
    }
    __syncthreads();
    if (!live) return;

    v8f c0 = {}, c1 = {}, c2 = {}, c3 = {};
    #pragma unroll
    for (int s = 0; s < 16; ++s) {
        const int kk = s * 4 + 2 * hi;
        v2f b;
        b.x = Bs[kk * D_ + 0 + colB];  b.y = Bs[(kk + 1) * D_ + 0 + colB];
        c0 = __builtin_amdgcn_wmma_f32_16x16x4_f32(false, a[s], false, b, (short)0, c0, false, false);
        b.x = Bs[kk * D_ + 16 + colB]; b.y = Bs[(kk + 1) * D_ + 16 + colB];
        c1 = __builtin_amdgcn_wmma_f32_16x16x4_f32(false, a[s], false, b, (short)0, c1, false, false);
        b.x = Bs[kk * D_ + 32 + colB]; b.y = Bs[(kk + 1) * D_ + 32 + colB];
        c2 = __builtin_amdgcn_wmma_f32_16x16x4_f32(false, a[s], false, b, (short)0, c2, false, false);
        b.x = Bs[kk * D_ + 48 + colB]; b.y = Bs[(kk + 1) * D_ + 48 + colB];
        c3 = __builtin_amdgcn_wmma_f32_16x16x4_f32(false, a[s], false, b, (short)0, c3, false, false);
    }

    if (idx == 0) {
        #pragma unroll
        for (int r = 0; r < 8; ++r) {
            const int bb = r + hi * 8;   // batch
            float* dst = T0 + (size_t)(bb * 128 + (i - 128)) * D_ + colB;
            dst[0] = c0[r]; dst[16] = c1[r]; dst[32] = c2[r]; dst[48] = c3[r];
        }
    } else {
        const int j = i + idx - 128;     // 0..127 by construction
        #pragma unroll
        for (int r = 0; r < 8; ++r) {
            const int bb = r + hi * 8;
            float* dst = priors + (size_t)((bb * J_ + j) * I_ + i) * D_ + colB;
            dst[0] = c0[r]; dst[16] = c1[r]; dst[32] = c2[r]; dst[48] = c3[r];
        }
    }
}

// ---------------------------------------------------------------------------
// broadcast clipped-region priors: for i in [128,255], j <= i-128:
//   priors[b,j,i,:] = T0[b,i-128,:]
// grid (128, 16), block 64 (p)
// ---------------------------------------------------------------------------
__global__ void k_priors_bcast(const float* __restrict__ T0, float* __restrict__ priors) {
    const int ii = blockIdx.x;           // i - 128
    const int b  = blockIdx.y;
    const int p  = threadIdx.x;
    const int i  = ii + 128;
    const float t = T0[(size_t)(b * 128 + ii) * D_ + p];
    float* base = priors + ((size_t)(b * J_) * I_ + i) * D_ + p;
    const int jmax = i - 128;            // inclusive
    for (int j = 0; j <= jmax; ++j)
        base[(size_t)j * I_ * D_] = t;
}

// ---------------------------------------------------------------------------
// mask + softmax over J; writes masked logits back and probs (+1e-6)
// one block per (b,i), 128 threads (thread = j)
// ---------------------------------------------------------------------------
__global__ void k_mask_softmax(const unsigned char* __restrict__ mask,
                               float* __restrict__ logits, float* __restrict__ probs) {
    const int bi = blockIdx.x;           // b*256 + i
    const int j  = threadIdx.x;
    const bool msk = mask[bi] != 0;
    float l = msk ? -10000.0f : logits[bi * J_ + j];
    logits[bi * J_ + j] = l;

    __shared__ float red[J_];
    red[j] = l;
    __syncthreads();
    for (int off = 64; off > 0; off >>= 1) {
        if (j < off) red[j] = fmaxf(red[j], red[j + off]);
        __syncthreads();
    }
    const float mx = red[0];
    __syncthreads();
    const float e = expf(l - mx);
    red[j] = e;
    __syncthreads();
    for (int off = 64; off > 0; off >>= 1) {
        if (j < off) red[j] += red[j + off];
        __syncthreads();
    }
    probs[bi * J_ + j] = e / red[0] + 1e-6f;
}

// ---------------------------------------------------------------------------
// s[b,j,:] = sum_i probs[b,i,j] * priors[b,j,i,:] ; v = squash(s)
// one block per (b,j), 256 threads: p = tid&63, i-chunk = tid>>6
// ---------------------------------------------------------------------------
__global__ void k_route_v(const float* __restrict__ priors, const float* __restrict__ probs,
                          float* __restrict__ v) {
    const int bj = blockIdx.x;           // b*128 + j
    const int b  = bj >> 7;
    const int j  = bj & 127;
    const int tid = threadIdx.x;
    const int p  = tid & 63;
    const int chunk = tid >> 6;

    const float* pb  = priors + (size_t)bj * I_ * D_;
    const float* prb = probs + (size_t)(b * I_) * J_ + j;

    float acc = 0.0f;
    const int i0 = chunk * 64;
    #pragma unroll 4
    for (int i = i0; i < i0 + 64; ++i)
        acc += prb[i * J_] * pb[(size_t)i * D_ + p];

    __shared__ float lds[256];
    __shared__ float sqs;
    lds[tid] = acc;
    __syncthreads();
    if (tid < 64)
        lds[tid] = lds[tid] + lds[tid + 64] + lds[tid + 128] + lds[tid + 192];
    __syncthreads();
    if (tid == 0) {
        float sq = 0.0f;
        for (int q = 0; q < 64; ++q) sq += lds[q] * lds[q];
        sqs = sq;
    }
    __syncthreads();
    if (tid < 64) {
        const float sq = sqs;
        const float scale = sq / (1.0f + sq);
        v[bj * D_ + tid] = scale * lds[tid] / (sqrtf(sq) + 1e-4f);
    }
}

// ---------------------------------------------------------------------------
// logits[b,i,j] += dot(priors[b,j,i,:], v[b,j,:]) / 8
// one wave per (b,i,j); wg = (b*128 + j)*256 + i so priors reads are coalesced
// ---------------------------------------------------------------------------
__global__ void k_route_delta(const float* __restrict__ priors, const float* __restrict__ v,
                              float* __restrict__ logits) {
    const int wg   = blockIdx.x * 8 + (threadIdx.x >> 5);
    const int lane = threadIdx.x & 31;
    const int i  = wg & 255;
    const int bj = wg >> 8;              // b*128 + j
    const int b  = bj >> 7;
    const int j  = bj & 127;

    const float* pb = priors + (size_t)wg * D_;
    const float* vb = v + bj * D_;
    float acc = pb[lane] * vb[lane] + pb[lane + 32] * vb[lane + 32];
    #pragma unroll
    for (int off = 16; off > 0; off >>= 1)
        acc += __shfl_xor(acc, off, 32);
    if (lane == 0)
        logits[(b * I_ + i) * J_ + j] += acc * 0.125f;
}

// ---------------------------------------------------------------------------
// out = v @ W_out + b_out            [2048 x 64] * [64 x 512]
// grid (128, 8), block 128: wave handles N-tile (blockIdx.y*4 + wave)
// ---------------------------------------------------------------------------
__global__ void k_out_gemm(const float* __restrict__ V, const float* __restrict__ W,
                           const float* __restrict__ bias, float* __restrict__ out) {
    const int lane = threadIdx.x & 31;
    const int wave = threadIdx.x >> 5;
    const int m    = lane & 15;
    const int hi   = lane >> 4;
    const int m0   = blockIdx.x * 16;
    const int n0   = (blockIdx.y * 4 + wave) * 16;
    const int colB = lane & 15;

    v8f c = {};
    #pragma unroll
    for (int k = 0; k < D_; k += 4) {
        const int kk = k + 2 * hi;
        v2f a = *(const v2f*)(V + (m0 + m) * D_ + kk);
        v2f b;
        b.x = W[kk * DOUT + n0 + colB];
        b.y = W[(kk + 1) * DOUT + n0 + colB];
        c = __builtin_amdgcn_wmma_f32_16x16x4_f32(false, a, false, b,
                                                  (short)0, c, false, false);
    }
    #pragma unroll
    for (int r = 0; r < 8; ++r) {
        const int Row = r + hi * 8;
        const int Col = lane & 15;
        out[(size_t)(m0 + Row) * DOUT + n0 + Col] = c[r] + bias[n0 + Col];
    }
}

// ---------------------------------------------------------------------------
extern "C" void kernel_launch(void* const* d_in, const int* in_sizes, int n_in,
                              void* d_out, int out_size, void* d_ws, size_t ws_size,
                              hipStream_t stream) {
    const float*         inputs_u = (const float*)d_in[0];
    const unsigned char* mask     = (const unsigned char*)d_in[1];   // jnp bool -> 1 byte
    // d_in[2] = num_out_caps (always 128, hardcoded)
    const float* W_in  = (const float*)d_in[3];
    const float* b_in  = (const float*)d_in[4];
    const float* W_out = (const float*)d_in[5];
    const float* b_out = (const float*)d_in[6];
    const float* route = (const float*)d_in[7];

    float* out   = (float*)d_out;                 // [16,128,512]
    float* probs = out + (size_t)B_ * J_ * DOUT;  // [16,256,128]

    float* ws     = (float*)d_ws;
    float* U      = ws;                                       // 4096*64
    float* priors = U + (size_t)B_ * I_ * D_;                 // 16*128*256*64
    float* v      = priors + (size_t)B_ * J_ * I_ * D_;       // 16*128*64
    float* logits = v + (size_t)B_ * J_ * D_;                 // 16*256*128
    float* T0     = logits + (size_t)B_ * I_ * J_;            // 16*128*64

    const int nLog = B_ * I_ * J_;
    k_zero<<<(nLog + 255) / 256, 256, 0, stream>>>(logits, nLog);

    k_u_gemm<<<(B_ * I_) / 16, 128, 0, stream>>>(inputs_u, W_in, b_in, U);

    k_priors_diag<<<dim3(8, 256), 512, 0, stream>>>(U, route, priors, T0);
    k_priors_bcast<<<dim3(128, B_), D_, 0, stream>>>(T0, priors);

    for (int it = 0; it < 3; ++it) {
        k_mask_softmax<<<B_ * I_, J_, 0, stream>>>(mask, logits, probs);
        k_route_v<<<B_ * J_, 256, 0, stream>>>(priors, probs, v);
        if (it < 2)
            k_route_delta<<<(B_ * I_ * J_) / 8, 256, 0, stream>>>(priors, v, logits);
    }

    k_out_gemm<<<dim3((B_ * J_) / 16, DOUT / 64), 128, 0, stream>>>(v, W_out, b_out, out);
}